// GraphLayer_32693291057755
// MI455X (gfx1250) — compile-verified
//
#include <hip/hip_runtime.h>
#include <math.h>

#define N_NODES 10000
#define N_EDGES 640000
#define IN_DIM  128
#define OUT_DIM 128

typedef __attribute__((ext_vector_type(2))) float v2f;
typedef __attribute__((ext_vector_type(8))) float v8f;

// ---------------------------------------------------------------------------
// 0) init: out = 0, m = -inf, den = 0
// ---------------------------------------------------------------------------
__global__ void gat_init(float* __restrict__ out, float* __restrict__ m,
                         float* __restrict__ den) {
    int i = blockIdx.x * 256 + threadIdx.x;
    if (i < N_NODES * OUT_DIM) out[i] = 0.0f;
    if (i < N_NODES) { m[i] = -INFINITY; den[i] = 0.0f; }
}

// ---------------------------------------------------------------------------
// 1) z = h @ W_fc^T via V_WMMA_F32_16X16X4_F32
//    block = 256 threads = 8 waves; wave w computes tile (blockIdx.x, w)
//    A = h[16x4] tile, B[k][n] = W_fc[n][k] -> both read contiguous float2
//    along IN_DIM. 10000 = 625*16, no remainder handling needed.
// ---------------------------------------------------------------------------
__global__ void gat_gemm_wmma(const float* __restrict__ h,
                              const float* __restrict__ W,
                              float* __restrict__ z) {
    const int lane  = threadIdx.x & 31;
    const int wave  = threadIdx.x >> 5;       // N-tile within block (0..7)
    const int tileM = blockIdx.x * 16;        // 625 blocks
    const int tileN = wave * 16;
    const int r  = lane & 15;                 // row (A) / col (B)
    const int kk = (lane >> 4) << 1;          // 0 or 2: K sub-pair

    const float* __restrict__ arow = h + (size_t)(tileM + r) * IN_DIM;
    const float* __restrict__ brow = W + (size_t)(tileN + r) * IN_DIM;

    v8f c = {};
#pragma unroll
    for (int k = 0; k < IN_DIM; k += 4) {
        v2f a = *(const v2f*)(arow + k + kk);   // even index -> 8B aligned
        v2f b = *(const v2f*)(brow + k + kk);
        c = __builtin_amdgcn_wmma_f32_16x16x4_f32(
                /*neg_a=*/false, a, /*neg_b=*/false, b,
                /*c_mod=*/(short)0, c, /*reuse_a=*/false, /*reuse_b=*/false);
    }

    const int rowBase = tileM + ((lane >> 4) << 3);  // lanes 16-31 -> +8
#pragma unroll
    for (int v = 0; v < 8; ++v)
        z[(size_t)(rowBase + v) * OUT_DIM + tileN + r] = c[v];
}

// ---------------------------------------------------------------------------
// 2) e = z @ a_src, f = z @ a_dst   (one wave per node, lane owns 4 dims)
// ---------------------------------------------------------------------------
__global__ void gat_ef(const float* __restrict__ z,
                       const float* __restrict__ W_attn,
                       float* __restrict__ e, float* __restrict__ f) {
    const int node = blockIdx.x * 8 + (threadIdx.x >> 5);
    const int lane = threadIdx.x & 31;
    if (node >= N_NODES) return;
    const float4 zz = *(const float4*)(z + (size_t)node * OUT_DIM + lane * 4);
    const float4 as = *(const float4*)(W_attn + lane * 4);
    const float4 ad = *(const float4*)(W_attn + OUT_DIM + lane * 4);
    float pe = zz.x*as.x + zz.y*as.y + zz.z*as.z + zz.w*as.w;
    float pf = zz.x*ad.x + zz.y*ad.y + zz.z*ad.z + zz.w*ad.w;
#pragma unroll
    for (int off = 16; off > 0; off >>= 1) {
        pe += __shfl_down(pe, off, 32);
        pf += __shfl_down(pf, off, 32);
    }
    if (lane == 0) { e[node] = pe; f[node] = pf; }
}

// float atomic max via signed-max / unsigned-min punning (monotonic encoding)
__device__ inline void atomic_max_float(float* addr, float val) {
    if (val >= 0.0f)
        atomicMax((int*)addr, __float_as_int(val));
    else
        atomicMin((unsigned int*)addr, (unsigned int)__float_as_int(val));
}

// ---------------------------------------------------------------------------
// 3) score[e] = e[src] + f[dst]; m[dst] = max(m[dst], score)
// ---------------------------------------------------------------------------
__global__ void gat_score(const int* __restrict__ src, const int* __restrict__ dst,
                          const float* __restrict__ e, const float* __restrict__ f,
                          float* __restrict__ score, float* __restrict__ m) {
    int i = blockIdx.x * 256 + threadIdx.x;
    if (i >= N_EDGES) return;
    const int d = dst[i];
    const float s = e[src[i]] + f[d];
    score[i] = s;
    atomic_max_float(&m[d], s);
}

// ---------------------------------------------------------------------------
// 4) m = isfinite(m) ? m : 0
// ---------------------------------------------------------------------------
__global__ void gat_finalize_m(float* __restrict__ m) {
    int i = blockIdx.x * 256 + threadIdx.x;
    if (i < N_NODES) {
        float v = m[i];
        m[i] = __builtin_isfinite(v) ? v : 0.0f;
    }
}

// ---------------------------------------------------------------------------
// 5) ex = exp(score - m[dst]); den[dst] += ex  (score buffer reused for ex)
// ---------------------------------------------------------------------------
__global__ void gat_exp(const int* __restrict__ dst, const float* __restrict__ m,
                        float* __restrict__ score, float* __restrict__ den) {
    int i = blockIdx.x * 256 + threadIdx.x;
    if (i >= N_EDGES) return;
    const int d = dst[i];
    const float ex = expf(score[i] - m[d]);
    score[i] = ex;
    atomicAdd(&den[d], ex);
}

// ---------------------------------------------------------------------------
// 6) out[dst] += (ex/den[dst]) * z[src]  (one wave per edge, float4 per lane)
// ---------------------------------------------------------------------------
__global__ void gat_aggregate(const int* __restrict__ src, const int* __restrict__ dst,
                              const float* __restrict__ ex, const float* __restrict__ den,
                              const float* __restrict__ z, float* __restrict__ out) {
    const int edge = blockIdx.x * 8 + (threadIdx.x >> 5);
    if (edge >= N_EDGES) return;
    const int lane = threadIdx.x & 31;
    const int s = src[edge];
    const int d = dst[edge];
    const float dn = den[d];
    const float alpha = ex[edge] / (dn > 0.0f ? dn : 1.0f);
    const float4 zz = *(const float4*)(z + (size_t)s * OUT_DIM + lane * 4);
    float* o = out + (size_t)d * OUT_DIM + lane * 4;
    atomicAdd(o + 0, alpha * zz.x);
    atomicAdd(o + 1, alpha * zz.y);
    atomicAdd(o + 2, alpha * zz.z);
    atomicAdd(o + 3, alpha * zz.w);
}

// ---------------------------------------------------------------------------
extern "C" void kernel_launch(void* const* d_in, const int* in_sizes, int n_in,
                              void* d_out, int out_size, void* d_ws, size_t ws_size,
                              hipStream_t stream) {
    const float* h      = (const float*)d_in[0];
    const int*   src    = (const int*)  d_in[1];
    const int*   dst    = (const int*)  d_in[2];
    const float* W_fc   = (const float*)d_in[3];
    const float* W_attn = (const float*)d_in[4];
    float* out = (float*)d_out;

    // workspace layout (floats); all offsets 16B-aligned
    float* z     = (float*)d_ws;                    // N_NODES*OUT_DIM = 1,280,000
    float* score = z + (size_t)N_NODES * OUT_DIM;   // N_EDGES         =   640,000
    float* m     = score + N_EDGES;                 // N_NODES
    float* den   = m + N_NODES;                     // N_NODES
    float* e     = den + N_NODES;                   // N_NODES
    float* f     = e + N_NODES;                     // N_NODES

    const int initN = N_NODES * OUT_DIM;
    gat_init<<<(initN + 255) / 256, 256, 0, stream>>>(out, m, den);

    gat_gemm_wmma<<<N_NODES / 16, 256, 0, stream>>>(h, W_fc, z);

    gat_ef<<<(N_NODES + 7) / 8, 256, 0, stream>>>(z, W_attn, e, f);

    gat_score<<<(N_EDGES + 255) / 256, 256, 0, stream>>>(src, dst, e, f, score, m);

    gat_finalize_m<<<(N_NODES + 255) / 256, 256, 0, stream>>>(m);

    gat_exp<<<(N_EDGES + 255) / 256, 256, 0, stream>>>(dst, m, score, den);

    gat_aggregate<<<(N_EDGES + 7) / 8, 256, 0, stream>>>(src, dst, score, den, z, out);
}